// Unfold_45243185496328
// MI455X (gfx1250) — compile-verified
//
#include <hip/hip_runtime.h>

// ---------------------------------------------------------------------------
// 3D unfold (im2col, K=3, same padding), MI455X / gfx1250.
// Zero-FLOP, store-bound op: 16 MiB read (L2-resident), 432 MiB written.
// Strategy: per-block async-stage the 9 needed input rows into LDS
// (GLOBAL_LOAD_ASYNC_TO_LDS_B128 / ASYNCcnt), then fan out 27 shifted rows
// with nontemporal B128 stores so the output stream doesn't pollute L2.
// ---------------------------------------------------------------------------

#define BD 2
#define CD 16
#define DD 32
#define HD 64
#define WD 64
#define S27 27   // K^3

typedef float f4 __attribute__((ext_vector_type(4)));
// Exact pointee type the async-LDS builtin expects (per hipcc diagnostic):
// 16-byte int vector; global side is AS1, LDS side is AS3.
typedef int vi4 __attribute__((vector_size(16)));
typedef __attribute__((address_space(1))) vi4* gvi4_ptr;
typedef __attribute__((address_space(3))) vi4* lvi4_ptr;

#ifndef __has_builtin
#define __has_builtin(x) 0
#endif

__global__ __launch_bounds__(256) void Unfold3D_gfx1250_kernel(
    const float* __restrict__ x, float* __restrict__ out) {
  // 9 input rows (dz,dy in {-1,0,1}) of 64 floats each.
  __shared__ float smem[9 * WD];

  const int bid = blockIdx.x;
  const int y = bid & (HD - 1);
  const int z = (bid >> 6) & (DD - 1);
  const int c = (bid >> 11) & (CD - 1);
  const int b = bid >> 15;
  const int tid = threadIdx.x;

  // ---- Load phase: 9 rows x 16 float4 chunks = 144 lanes of work ----------
  if (tid < 144) {
    const int r = tid >> 4;   // 0..8  : (dz*3 + dy)
    const int q = tid & 15;   // 0..15 : float4 chunk within row
    const int sz = z + (r / 3) - 1;
    const int sy = y + (r % 3) - 1;
    float* lrow = &smem[r * WD + q * 4];
    if ((unsigned)sz < (unsigned)DD && (unsigned)sy < (unsigned)HD) {
      const float* grow =
          x + (((size_t)(b * CD + c) * DD + sz) * HD + sy) * WD + q * 4;
#if __has_builtin(__builtin_amdgcn_global_load_async_to_lds_b128)
      __builtin_amdgcn_global_load_async_to_lds_b128(
          (gvi4_ptr)grow, (lvi4_ptr)lrow, /*offset=*/0, /*cpol=*/0);
#else
      *(f4*)lrow = *(const f4*)grow;
#endif
    } else {
      f4 zero = {0.f, 0.f, 0.f, 0.f};
      *(f4*)lrow = zero;  // padding rows
    }
  }

  // Drain this wave's outstanding async LDS loads, then publish to the block.
#if __has_builtin(__builtin_amdgcn_s_wait_asynccnt)
  __builtin_amdgcn_s_wait_asynccnt(0);
#else
  asm volatile("s_wait_asynccnt 0" ::: "memory");
#endif
  __syncthreads();

  // ---- Store phase: 27 shifted rows, 16 lanes (float4 each) per row -------
  const int rg = tid >> 4;  // 0..15: which output row this lane group handles
  const int q = tid & 15;   // float4 chunk within the 64-wide row
  const size_t obase = (size_t)(b * CD + c) * S27;

  for (int s = rg; s < S27; s += 16) {
    const int dz = s / 9;
    const int dy = (s / 3) % 3;
    const int dx = s % 3;
    const float* lrow = &smem[(dz * 3 + dy) * WD];

    f4 v;
#pragma unroll
    for (int j = 0; j < 4; ++j) {
      const int xx = q * 4 + j + dx - 1;  // shifted source column
      v[j] = ((unsigned)xx < (unsigned)WD) ? lrow[xx] : 0.f;
    }

    float* orow =
        out + (((obase + s) * DD + z) * HD + y) * WD + q * 4;
    // Output is write-once (432 MiB): nontemporal so L2 keeps the input hot.
    __builtin_nontemporal_store(v, (f4*)orow);
  }
}

extern "C" void kernel_launch(void* const* d_in, const int* in_sizes, int n_in,
                              void* d_out, int out_size, void* d_ws,
                              size_t ws_size, hipStream_t stream) {
  (void)in_sizes; (void)n_in; (void)out_size; (void)d_ws; (void)ws_size;
  const float* x = (const float*)d_in[0];
  float* out = (float*)d_out;

  const int nblocks = BD * CD * DD * HD;  // 2*16*32*64 = 65536
  Unfold3D_gfx1250_kernel<<<nblocks, 256, 0, stream>>>(x, out);
}